// RelationLoss_7687991460410
// MI455X (gfx1250) — compile-verified
//
#include <hip/hip_runtime.h>
#include <hip/hip_bf16.h>

typedef __attribute__((ext_vector_type(2))) float v2f;
typedef __attribute__((ext_vector_type(8))) float v8f;

// One wave (32 threads) per channel.
// Lane mapping: half = lane>>4 selects K pair {0,1} vs {2,3};
//               m = lane&15 is the matrix row; g = m>>2 picks one of 4
//               channel segments; v = m&3 picks vector (e, x, t, 1).
// A[m,k] = vec_v[ g*seg + kk + k ].  B is the mirror layout with identical
// per-lane content, so the same v2f feeds both operands and
// D = A x B accumulates the per-segment Gram matrices of (e,x,t,1) in the
// 4x4 diagonal blocks of the 16x16 f32 accumulator — exact f32 math.
__global__ __launch_bounds__(32)
void relation_moments_wmma(const float* __restrict__ x,
                           const float* __restrict__ st,
                           float* __restrict__ ch_sum,
                           int N) {
  const int c    = blockIdx.x;
  const int lane = threadIdx.x;
  const int half = lane >> 4;   // 0: K=0,1   1: K=2,3
  const int m    = lane & 15;   // matrix row / col id
  const int g    = m >> 2;      // segment 0..3
  const int v    = m & 3;       // 0=e, 1=x, 2=t, 3=ones
  const int seg  = N >> 2;      // 256 for N=1024

  const float* xc = x  + (size_t)c * N;
  const float* tc = st + (size_t)c * N;

  v8f acc = {};

  const int base = g * seg + half * 2;
  for (int kk = 0; kk < seg; kk += 4) {
    const int i0 = base + kk;                      // even -> 8B aligned
    const float2 xv = *(const float2*)(xc + i0);   // global_load_b64
    const float2 tv = *(const float2*)(tc + i0);
    // branchless vector select keeps EXEC all-1s for the WMMA
    const float e0 = xv.x * xv.x - tv.x * tv.x;
    const float e1 = xv.y * xv.y - tv.y * tv.y;
    float a0 = (v == 0) ? e0 : ((v == 1) ? xv.x : ((v == 2) ? tv.x : 1.0f));
    float a1 = (v == 0) ? e1 : ((v == 1) ? xv.y : ((v == 2) ? tv.y : 1.0f));
    v2f ab;
    ab.x = a0;
    ab.y = a1;
    // D = A*B + C, full f32 precision; accumulator chained through C.
    acc = __builtin_amdgcn_wmma_f32_16x16x4_f32(
        /*neg_a=*/false, ab, /*neg_b=*/false, ab,
        /*c_mod=*/(short)0, acc, /*reuse_a=*/false, /*reuse_b=*/false);
  }

  // Spill the 16x16 accumulator to LDS.
  // C/D layout: VGPR r, lanes 0-15 -> D[r, lane]; lanes 16-31 -> D[8+r, lane-16]
  __shared__ float Dm[16][16];
  const int n_col  = lane & 15;
  const int m_base = half * 8;
#pragma unroll
  for (int r = 0; r < 8; ++r)
    Dm[m_base + r][n_col] = acc[r];
  __syncthreads();

  if (lane == 0) {
    // Sum the 4 segment Gram matrices (diagonal 4x4 blocks). Gram is
    // symmetric, so either triangle works.
    float Gee = 0.f, Ge1 = 0.f, Ges = 0.f, Get = 0.f;
    float Gs1 = 0.f, Gt1 = 0.f, Gss = 0.f, Gtt = 0.f, Gst = 0.f;
#pragma unroll
    for (int gg = 0; gg < 4; ++gg) {
      const int b = 4 * gg;
      Gee += Dm[b + 0][b + 0];   // sum e^2
      Ges += Dm[b + 0][b + 1];   // sum e*x
      Get += Dm[b + 0][b + 2];   // sum e*t
      Ge1 += Dm[b + 0][b + 3];   // sum e
      Gss += Dm[b + 1][b + 1];   // sum x^2
      Gst += Dm[b + 1][b + 2];   // sum x*t
      Gs1 += Dm[b + 1][b + 3];   // sum x
      Gtt += Dm[b + 2][b + 2];   // sum t^2
      Gt1 += Dm[b + 2][b + 3];   // sum t
    }
    const float Nf = (float)N;
    // sum_{i,j} ((x_j-x_i)^2 - (t_j-t_i)^2)^2 in closed form:
    const float chsum =
        2.0f * Nf * Gee + 2.0f * Ge1 * Ge1
      - 8.0f * (Ges * Gs1 - Get * Gt1)
      + 4.0f * (Gss * Gss - 2.0f * Gst * Gst + Gtt * Gtt);
    ch_sum[c] = chsum;
  }
}

// Fixed-order tree reduction of the per-channel sums -> deterministic output.
__global__ __launch_bounds__(128)
void relation_finalize(const float* __restrict__ ch_sum,
                       float* __restrict__ out,
                       int C, float scale) {
  __shared__ float s[128];
  const int t = threadIdx.x;
  float acc = 0.f;
  for (int i = t; i < C; i += 128) acc += ch_sum[i];
  s[t] = acc;
  __syncthreads();
#pragma unroll
  for (int stride = 64; stride > 0; stride >>= 1) {
    if (t < stride) s[t] += s[t + stride];
    __syncthreads();
  }
  if (t == 0) out[0] = s[0] * scale;
}

extern "C" void kernel_launch(void* const* d_in, const int* in_sizes, int n_in,
                              void* d_out, int out_size, void* d_ws, size_t ws_size,
                              hipStream_t stream) {
  const float* x  = (const float*)d_in[0];
  const float* st = (const float*)d_in[1];
  const int N = 1024;                 // reference: shape (C=128, N=1024)
  const int C = in_sizes[0] / N;

  float* ch = (float*)d_ws;           // C floats of scratch

  relation_moments_wmma<<<C, 32, 0, stream>>>(x, st, ch, N);

  const float scale =
      (float)(1.0 /* LOSS_WEIGHT */ / ((double)C * (double)N * (double)N));
  relation_finalize<<<1, 128, 0, stream>>>(ch, (float*)d_out, C, scale);
}